// LoRAGroupedLinear_51041391345752
// MI455X (gfx1250) — compile-verified
//
#include <hip/hip_runtime.h>
#include <stdint.h>

typedef __attribute__((ext_vector_type(2))) float v2f;
typedef __attribute__((ext_vector_type(8))) float v8f;
typedef unsigned int u32x4 __attribute__((ext_vector_type(4)));
typedef unsigned int u32x8 __attribute__((ext_vector_type(8)));

#define G_GROUPS 8
#define DIN 1024
#define DOUT 1024
#define RNK 16
#define NTOK 8192
#define LORA_SCALE 2.0f
#define BK 64
#define WTP 136   // W_base LDS row stride in dwords: 128 + 8 (TDM pad) -> conflict-free

// Low 32 bits of a generic/shared pointer == LDS byte offset (flat LDS aperture).
__device__ __forceinline__ unsigned lds_off(const void* p) {
    return (unsigned)(uintptr_t)p;
}

// Issue one 2D TDM tile load: tile_d1 rows x tile_d0 elems (4-byte elements)
// from a row-major tensor with row stride stride0 (elems), into LDS at lds_addr.
// pad_ctl carries the pad_enable/pad_interval/pad_amount bits of D# group1 word0.
__device__ __forceinline__ void tdm_load_2d(
    unsigned lds_addr, unsigned long long gaddr,
    unsigned tensor_d0, unsigned tensor_d1,
    unsigned tile_d0, unsigned tile_d1,
    unsigned stride0, unsigned pad_ctl)
{
    u32x4 d0;
    d0[0] = 1u;                                            // count=1 (valid), user mode
    d0[1] = lds_addr;                                      // lds_addr [63:32]
    d0[2] = (unsigned)(gaddr & 0xFFFFFFFFull);             // global_addr lo
    d0[3] = (unsigned)((gaddr >> 32) & 0x01FFFFFFull)      // global_addr [56:32]
            | 0x80000000u;                                 // type=2 (bits 127:126)
    u32x8 d1;
    d1[0] = (2u << 16) | pad_ctl;                          // data_size=4B, wg_mask=0
    d1[1] = (tensor_d0 & 0xFFFFu) << 16;                   // tensor_dim0 [79:48]
    d1[2] = (tensor_d0 >> 16) | ((tensor_d1 & 0xFFFFu) << 16); // tensor_dim1 [111:80]
    d1[3] = (tensor_d1 >> 16) | (tile_d0 << 16);           // tile_dim0 [127:112]
    d1[4] = tile_d1;                                       // tile_dim1; tile_dim2=0 (2D)
    d1[5] = stride0;                                       // tensor_dim0_stride lo32
    d1[6] = 0u;                                            // stride0 hi / stride1 lo
    d1[7] = 0u;                                            // stride1 hi
    asm volatile("tensor_load_to_lds %0, %1" :: "s"(d0), "s"(d1) : "memory");
}

// Block: 256 threads = 8 waves (wave32). Block tile: 16 rows x 128 cols.
// Wave w owns cols [blockIdx.x*128 + w*16, +16). K-loop over DIN in BK chunks,
// W_base/W_A chunks streamed into LDS by the Tensor Data Mover (double-buffered).
__global__ __launch_bounds__(256) void lora_grouped_gemm_f32wmma(
    const float* __restrict__ x,      // [NTOK, DIN]
    const int*   __restrict__ xg,     // [NTOK] sorted group ids
    const float* __restrict__ Wb,     // [G, DIN, DOUT]
    const float* __restrict__ Wa,     // [G, DIN, RNK]
    const float* __restrict__ Wl,     // [G, RNK, DOUT]
    float*       __restrict__ out)    // [NTOK, DOUT]
{
    __shared__ float xs[16][BK + 4];      // masked x tile (stride 68 dodges bank conflicts)
    __shared__ float at[8][16][17];       // per-wave scaled LoRA 'a' tile
    __shared__ float wt[2][BK][WTP];      // TDM-staged W_base tiles (double buffer)
    __shared__ float wl[2][BK][RNK];      // TDM-staged W_A tiles   (double buffer)

    const int lane  = threadIdx.x & 31;
    const int wave  = threadIdx.x >> 5;
    const int mrow  = lane & 15;            // M-row (A) / N-col (B,C,D) index
    const int khalf = (lane >> 4) << 1;     // 0 for lanes 0-15, 2 for lanes 16-31
    const int chalf = (lane >> 4) << 3;     // C/D: M base 0 or 8
    const int wcol  = (wave << 4) | mrow;   // column within the 128-wide block tile

    const int row0    = blockIdx.y * 16;
    const int colBase = blockIdx.x * 128;
    const int col0    = colBase + (wave << 4);

    // sorted groups => tile spans [gFirst, gLast]; usually gFirst == gLast
    const int gFirst = xg[row0];
    const int gLast  = xg[row0 + 15];

    // x-tile loader coords: thread t -> row t/16, 4 floats at col (t%16)*4
    const int lr = threadIdx.x >> 4;
    const int lc = (threadIdx.x & 15) << 2;
    const int lrGroup = xg[row0 + lr];
    const float* xRowPtr = x + (size_t)(row0 + lr) * DIN;

    // TDM pad control for the W_base tile: every 128 dwords insert 8 dwords.
    const unsigned PAD_WB = (1u << 20) | (6u << 22) | (7u << 25);

    v8f acc = {};   // fp32 16x16 output accumulator (base + scaled adapter)

    for (int g = gFirst; g <= gLast; ++g) {
        v8f accA = {};  // a = x_masked @ W_A[g], 16x16 (N-dim == RNK)
        const float* WbG = Wb + (size_t)g * DIN * DOUT;
        const float* WaG = Wa + (size_t)g * DIN * RNK;
        const float rowOn = (lrGroup == g) ? 1.0f : 0.0f;

        // Prologue: stream chunk 0 into buffer 0.
        if (wave == 0) {
            tdm_load_2d(lds_off(&wt[0][0][0]),
                        (unsigned long long)(uintptr_t)(WbG + colBase),
                        DOUT, DIN, 128, BK, DOUT, PAD_WB);
            tdm_load_2d(lds_off(&wl[0][0][0]),
                        (unsigned long long)(uintptr_t)WaG,
                        RNK, DIN, RNK, BK, RNK, 0u);
        }

        for (int k0 = 0, it = 0; k0 < DIN; k0 += BK, ++it) {
            const int buf = it & 1;
            __syncthreads();   // previous chunk fully consumed; xs free

            // Stage masked x chunk: 16 x BK floats (zero rows not in group g).
            {
                const float* p = xRowPtr + k0 + lc;
                xs[lr][lc + 0] = p[0] * rowOn;
                xs[lr][lc + 1] = p[1] * rowOn;
                xs[lr][lc + 2] = p[2] * rowOn;
                xs[lr][lc + 3] = p[3] * rowOn;
            }

            if (wave == 0) {
                if (k0 + BK < DIN) {
                    // Stream next chunk into the other buffer (freed at barrier above),
                    // then wait until this chunk's two tensor ops have retired.
                    const int nb = buf ^ 1;
                    tdm_load_2d(lds_off(&wt[nb][0][0]),
                                (unsigned long long)(uintptr_t)
                                    (WbG + (size_t)(k0 + BK) * DOUT + colBase),
                                DOUT, DIN, 128, BK, DOUT, PAD_WB);
                    tdm_load_2d(lds_off(&wl[nb][0][0]),
                                (unsigned long long)(uintptr_t)
                                    (WaG + (size_t)(k0 + BK) * RNK),
                                RNK, DIN, RNK, BK, RNK, 0u);
                    __builtin_amdgcn_s_wait_tensorcnt(2);
                } else {
                    __builtin_amdgcn_s_wait_tensorcnt(0);
                }
            }
            __syncthreads();   // TDM data + x tile visible to all waves

            #pragma unroll
            for (int kk = 0; kk < BK; kk += 4) {
                // A fragment (16x4 f32): lane holds row mrow, K = kk+khalf+{0,1}
                v2f a;
                a.x = xs[mrow][kk + khalf];
                a.y = xs[mrow][kk + khalf + 1];

                // B fragment from LDS-staged W_base tile
                v2f b;
                b.x = wt[buf][kk + khalf][wcol];
                b.y = wt[buf][kk + khalf + 1][wcol];
                acc = __builtin_amdgcn_wmma_f32_16x16x4_f32(
                    false, a, false, b, (short)0, acc, false, false);

                // B fragment from LDS-staged W_A tile (RNK == 16 fills N exactly)
                v2f ba;
                ba.x = wl[buf][kk + khalf][mrow];
                ba.y = wl[buf][kk + khalf + 1][mrow];
                accA = __builtin_amdgcn_wmma_f32_16x16x4_f32(
                    false, a, false, ba, (short)0, accA, false, false);
            }
        }

        // Convert accA (C-layout) -> A-layout via LDS; fold in alpha/r scale.
        __syncthreads();
        #pragma unroll
        for (int r = 0; r < 8; ++r)
            at[wave][r + chalf][mrow] = accA[r] * LORA_SCALE;
        __syncthreads();

        // out += a_scaled @ W_B[g]  (K = RNK = 16 -> 4 WMMA steps)
        const float* WlG = Wl + (size_t)g * RNK * DOUT;
        #pragma unroll
        for (int kk = 0; kk < RNK; kk += 4) {
            v2f a;
            a.x = at[wave][mrow][kk + khalf];
            a.y = at[wave][mrow][kk + khalf + 1];
            const float* bp = WlG + (size_t)(kk + khalf) * DOUT + col0 + mrow;
            v2f b;
            b.x = bp[0];
            b.y = bp[DOUT];
            acc = __builtin_amdgcn_wmma_f32_16x16x4_f32(
                false, a, false, b, (short)0, acc, false, false);
        }
    }

    // Store D: VGPR r -> row row0 + r + chalf, col col0 + mrow
    float* op = out + (size_t)(row0 + chalf) * DOUT + col0 + mrow;
    #pragma unroll
    for (int r = 0; r < 8; ++r)
        op[(size_t)r * DOUT] = acc[r];
}

extern "C" void kernel_launch(void* const* d_in, const int* in_sizes, int n_in,
                              void* d_out, int out_size, void* d_ws, size_t ws_size,
                              hipStream_t stream) {
    (void)in_sizes; (void)n_in; (void)d_ws; (void)ws_size; (void)out_size;
    const float* x  = (const float*)d_in[0];
    const int*   xg = (const int*)d_in[1];
    const float* Wb = (const float*)d_in[2];
    const float* Wa = (const float*)d_in[3];
    const float* Wl = (const float*)d_in[4];
    float* out = (float*)d_out;

    dim3 grid(DOUT / 128, NTOK / 16);
    dim3 block(256);
    lora_grouped_gemm_f32wmma<<<grid, block, 0, stream>>>(x, xg, Wb, Wa, Wl, out);
}